// VQ_35467839930710
// MI455X (gfx1250) — compile-verified
//
#include <hip/hip_runtime.h>

typedef float v2f __attribute__((ext_vector_type(2)));
typedef float v8f __attribute__((ext_vector_type(8)));
typedef int   v4i __attribute__((vector_size(16)));   // matches builtin proto

#define B_ 16
#define C_ 512
#define T_ 4096
#define G_ 2
#define K_ 2048
#define E_ 256
#define TILE_T 128        // tokens per block
#define LDS_STRIDE 264    // floats; multiple of 4 -> 16B-aligned rows
#define CB_BUF (16 * LDS_STRIDE)

#if defined(__HIP_DEVICE_COMPILE__) && __has_builtin(__builtin_amdgcn_global_load_async_to_lds_b128)
#define HAS_ASYNC_LDS 1
#else
#define HAS_ASYNC_LDS 0
#endif

#if HAS_ASYNC_LDS
__device__ __forceinline__ void async_copy16(const float* g, float* l) {
    // GLOBAL_LOAD_ASYNC_TO_LDS_B128: direct global->LDS, tracked by ASYNCcnt.
    // AS3 pointer value is the low-32-bit LDS byte offset (flat aperture rule).
    __builtin_amdgcn_global_load_async_to_lds_b128(
        (__attribute__((address_space(1))) v4i*)(unsigned long long)g,
        (__attribute__((address_space(3))) v4i*)(unsigned int)(unsigned long long)l,
        0, 0);
}
__device__ __forceinline__ void async_wait_all() {
#if __has_builtin(__builtin_amdgcn_s_wait_asynccnt)
    __builtin_amdgcn_s_wait_asynccnt(0);
#else
    asm volatile("s_wait_asynccnt 0x0" ::: "memory");
#endif
}
#endif

__global__ __launch_bounds__(256, 1)
void vq_argmin_gather_kernel(const float* __restrict__ x,
                             const float* __restrict__ cb,
                             float* __restrict__ outQ,
                             float* __restrict__ outIdx)
{
    // Double-buffered codebook tiles; same region reused as the gather
    // transpose buffer (32*LDS_STRIDE == 2*CB_BUF exactly).
    __shared__ float smem[2 * CB_BUF];
    __shared__ float esqS[2][16];
    __shared__ int   idxS[TILE_T];

    const int tid  = threadIdx.x;
    const int lane = tid & 31;
    const int w    = tid >> 5;      // wave 0..7
    const int l15  = lane & 15;
    const int hi   = lane >> 4;     // half-wave select

    const int bid   = blockIdx.x;
    const int tTile = bid & 31;            // T_/TILE_T == 32
    const int b     = (bid >> 5) & 15;
    const int g     = bid >> 9;
    const int t0    = tTile * TILE_T;

    const int row = tid >> 4;        // staging row 0..15
    const int col = (tid & 15) * 4;  // staging col base
    const int seg = tid & 15;        // |e|^2 segment

    // ---------------- A panel: 16 tokens x E=256 into registers ----------
    // token m = l15; this wave covers t = t0 + w*16 + m
    // x channel for (e,g) is c = e*2 + g  (reshape(B,E,G,T))
    const float* xbase = x + ((size_t)b * C_ + g) * T_ + t0 + w * 16 + l15;
    v2f areg[64];
#pragma unroll
    for (int kc = 0; kc < 64; ++kc) {
        const int e0 = kc * 4 + 2 * hi;
        v2f a;
        a.x = xbase[(size_t)(e0    ) * 2 * T_];
        a.y = xbase[(size_t)(e0 + 1) * 2 * T_];
        areg[kc] = a;
    }

    float bestS[8];
    int   bestI[8];
#pragma unroll
    for (int r = 0; r < 8; ++r) { bestS[r] = 3.0e38f; bestI[r] = 0; }

    const float* cbg = cb + (size_t)g * K_ * E_;

    // ---------------- prologue: stage tile 0 into buffer 0 ----------------
    {
        const float* cbt = cbg;   // tile 0
#if HAS_ASYNC_LDS
#pragma unroll
        for (int i = 0; i < 4; ++i) {
            const int e4 = col + i * 64;
            async_copy16(cbt + row * E_ + e4, &smem[row * LDS_STRIDE + e4]);
        }
#else
#pragma unroll
        for (int i = 0; i < 4; ++i) {
            const int e4 = col + i * 64;
            *reinterpret_cast<float4*>(&smem[row * LDS_STRIDE + e4]) =
                *reinterpret_cast<const float4*>(cbt + row * E_ + e4);
        }
#endif
        const float* p = cbt + row * E_ + seg * 16;
        float s = 0.f;
#pragma unroll
        for (int i = 0; i < 16; ++i) { const float v = p[i]; s += v * v; }
#pragma unroll
        for (int m2 = 8; m2 >= 1; m2 >>= 1)
            s += __shfl_xor(s, m2, 16);
        if (seg == 0) esqS[0][row] = s;
#if HAS_ASYNC_LDS
        async_wait_all();
#endif
        __syncthreads();
    }

    // ---------------- scan codebook: 128 tiles of 16 codes ---------------
    for (int kt = 0; kt < K_ / 16; ++kt) {
        const int p = kt & 1;
        const float* cbuf = smem + p * CB_BUF;

        // stage tile kt+1 into the idle buffer (overlaps the WMMAs below);
        // the idle buffer's readers finished at the end of iteration kt-1.
        if (kt + 1 < K_ / 16) {
            const float* cbn = cbg + (size_t)(kt + 1) * 16 * E_;
            float* nbuf = smem + (1 - p) * CB_BUF;
#if HAS_ASYNC_LDS
#pragma unroll
            for (int i = 0; i < 4; ++i) {
                const int e4 = col + i * 64;
                async_copy16(cbn + row * E_ + e4, &nbuf[row * LDS_STRIDE + e4]);
            }
#else
#pragma unroll
            for (int i = 0; i < 4; ++i) {
                const int e4 = col + i * 64;
                *reinterpret_cast<float4*>(&nbuf[row * LDS_STRIDE + e4]) =
                    *reinterpret_cast<const float4*>(cbn + row * E_ + e4);
            }
#endif
            // |e|^2 for tile kt+1 (reads L2 directly; overlaps compute)
            const float* pp = cbn + row * E_ + seg * 16;
            float s = 0.f;
#pragma unroll
            for (int i = 0; i < 16; ++i) { const float v = pp[i]; s += v * v; }
#pragma unroll
            for (int m2 = 8; m2 >= 1; m2 >>= 1)
                s += __shfl_xor(s, m2, 16);
            if (seg == 0) esqS[1 - p][row] = s;
        }

        // 16x16 dot tile: 64 x V_WMMA_F32_16X16X4_F32 over E=256
        v8f c = {};
#pragma unroll
        for (int kc = 0; kc < 64; ++kc) {
            const v2f bfrag = *reinterpret_cast<const v2f*>(
                &cbuf[l15 * LDS_STRIDE + kc * 4 + 2 * hi]);
            c = __builtin_amdgcn_wmma_f32_16x16x4_f32(
                    false, areg[kc], false, bfrag, (short)0, c, false, false);
        }

        // argmin score: |e|^2 - 2*dot  (|x|^2 constant per token, drops out)
        const float esq      = esqS[p][l15];
        const int   codeBase = kt * 16 + l15;
#pragma unroll
        for (int r = 0; r < 8; ++r) {
            const float score = esq - 2.0f * c[r];
            if (score < bestS[r]) { bestS[r] = score; bestI[r] = codeBase; }
        }

#if HAS_ASYNC_LDS
        async_wait_all();     // this wave's async stage of tile kt+1 landed
#endif
        __syncthreads();      // all waves done reading cbuf -> reusable
    }

    // cross-lane argmin over the 16 code-lanes per token (first-min ties)
#pragma unroll
    for (int r = 0; r < 8; ++r) {
        float s = bestS[r]; int idx = bestI[r];
#pragma unroll
        for (int m2 = 8; m2 >= 1; m2 >>= 1) {
            const float os = __shfl_xor(s, m2, 16);
            const int   oi = __shfl_xor(idx, m2, 16);
            if (os < s || (os == s && oi < idx)) { s = os; idx = oi; }
        }
        bestS[r] = s; bestI[r] = idx;
    }

    if (l15 == 0) {
#pragma unroll
        for (int r = 0; r < 8; ++r) {
            const int m  = r + 8 * hi;       // token row in C/D layout
            const int tl = w * 16 + m;
            idxS[tl] = bestI[r];
            outIdx[((size_t)g * B_ + b) * T_ + t0 + tl] = (float)bestI[r];
        }
    }
    __syncthreads();

    // -------- gather + transpose store: 4 chunks of 32 tokens ------------
    // smem reused as qS[32][LDS_STRIDE]
    for (int ch = 0; ch < 4; ++ch) {
#pragma unroll
        for (int j = 0; j < 4; ++j) {
            const int tl32 = w * 4 + j;
            const int idx  = idxS[ch * 32 + tl32];
            const float* rowp = cbg + (size_t)idx * E_;
#pragma unroll
            for (int pq = 0; pq < 2; ++pq) {
                const int e = pq * 128 + lane * 4;
                const float4 v = *reinterpret_cast<const float4*>(rowp + e);
                *reinterpret_cast<float4*>(&smem[tl32 * LDS_STRIDE + e]) = v;
            }
        }
        __syncthreads();
        {
            const int tl32 = tid & 31;
            const int eg   = tid >> 5;
            const int t    = t0 + ch * 32 + tl32;
#pragma unroll
            for (int i = 0; i < 32; ++i) {
                const int e = eg * 32 + i;
                outQ[((size_t)b * C_ + g * E_ + e) * T_ + t] =
                    smem[tl32 * LDS_STRIDE + e];
            }
        }
        __syncthreads();
    }
}

extern "C" void kernel_launch(void* const* d_in, const int* in_sizes, int n_in,
                              void* d_out, int out_size, void* d_ws, size_t ws_size,
                              hipStream_t stream) {
    const float* x  = (const float*)d_in[0];   // (B, C, T) fp32
    const float* cb = (const float*)d_in[1];   // (G, K, E) fp32
    float* outQ   = (float*)d_out;                         // (B, C, T)
    float* outIdx = outQ + (size_t)B_ * C_ * T_;           // (G, B, T) as float

    const dim3 grid(G_ * B_ * (T_ / TILE_T));  // 1024 blocks
    vq_argmin_gather_kernel<<<grid, 256, 0, stream>>>(x, cb, outQ, outIdx);
}